// SubgraphGNNEncoder_57054345560646
// MI455X (gfx1250) — compile-verified
//
#include <hip/hip_runtime.h>
#include <hip/hip_bf16.h>

typedef float v2f __attribute__((ext_vector_type(2)));
typedef float v8f __attribute__((ext_vector_type(8)));

#define HD 128            // hidden dim
#define LDST 132          // LDS row stride (floats): addr%64 = 4*row + k -> conflict-free b64
#define BN_EPS 1e-5f

// ---------------------------------------------------------------------------
// Plain WMMA f32 GEMM (node projection): C = A @ W + bias, no activation.
// 256 threads = 8 waves; wave w computes rows [blk*128 + 16w, +16), all 128 cols.
// ---------------------------------------------------------------------------
__global__ __launch_bounds__(256) void gemm_f32_wmma(
    const float* __restrict__ A, const float* __restrict__ W,
    const float* __restrict__ bias, float* __restrict__ C, int nrows)
{
    __shared__ float wt[HD * LDST];   // wt[n*LDST + k] = W[k*128 + n]

    const int t = threadIdx.x;
    for (int idx = t; idx < HD * HD; idx += 256) {
        int k = idx >> 7, n = idx & 127;
        wt[n * LDST + k] = W[idx];
    }
    __syncthreads();

    const int wave  = t >> 5;
    const int lane  = t & 31;
    const int m     = lane & 15;
    const int khalf = lane >> 4;
    const int rowbase = blockIdx.x * 128 + wave * 16;

    int arow = rowbase + m;
    if (arow >= nrows) arow = nrows - 1;
    const float* Ap = A + (size_t)arow * HD + 2 * khalf;

    v8f acc[8];
    #pragma unroll
    for (int j = 0; j < 8; ++j) {
        float b = bias[j * 16 + m];
        #pragma unroll
        for (int r = 0; r < 8; ++r) acc[j][r] = b;
    }

    #pragma unroll 2
    for (int kk = 0; kk < 32; ++kk) {
        v2f a = *(const v2f*)(Ap + kk * 4);
        const int kb = kk * 4 + 2 * khalf;
        #pragma unroll
        for (int j = 0; j < 8; ++j) {
            v2f b = *(const v2f*)&wt[(j * 16 + m) * LDST + kb];
            acc[j] = __builtin_amdgcn_wmma_f32_16x16x4_f32(
                false, a, false, b, (short)0, acc[j], false, false);
        }
    }

    // Epilogue: VGPR r holds row (rowbase + 8*khalf + r), col (16j + m).
    float* Cp = C + (size_t)(rowbase + khalf * 8) * HD + m;
    if (rowbase + 16 <= nrows) {               // uniform fast path: full tile
        #pragma unroll
        for (int j = 0; j < 8; ++j) {
            #pragma unroll
            for (int r = 0; r < 8; ++r) Cp[(size_t)r * HD + j * 16] = acc[j][r];
        }
    } else {                                   // partial last tile
        #pragma unroll
        for (int j = 0; j < 8; ++j) {
            #pragma unroll
            for (int r = 0; r < 8; ++r) {
                int row = rowbase + khalf * 8 + r;
                if (row < nrows) Cp[(size_t)r * HD + j * 16] = acc[j][r];
            }
        }
    }
}

// ---------------------------------------------------------------------------
// Fully fused 4-layer MLP (GINE "nn") + BN stats:
//   z0 = (1+eps)*h + agg                              (fused into stage-0 A load)
//   z  = relu(z @ W_s + b_s) for s=0..2, then z @ W_3 + b_3 (no relu)
// Inter-layer activations live in LDS (wave-private 16-row blocks); only the
// 64KB weight matrix is re-staged per stage. Last stage stores Z to global and
// accumulates BN column sum/sumsq via fp32 global atomics.
// ---------------------------------------------------------------------------
__global__ __launch_bounds__(256) void mlp4_f32_wmma(
    const float* __restrict__ h, const float* __restrict__ agg,
    const float* __restrict__ epsArr, int layer,
    const float* __restrict__ Wl, const float* __restrict__ bl,
    float* __restrict__ Z, int nrows,
    float* __restrict__ bnsum, float* __restrict__ bnsumsq)
{
    __shared__ float wt[HD * LDST];   // transposed weights for current stage
    __shared__ float at[HD * LDST];   // activation tile, row-major, stride LDST

    const int t = threadIdx.x;
    const int wave  = t >> 5;
    const int lane  = t & 31;
    const int m     = lane & 15;
    const int khalf = lane >> 4;
    const int rowbase = blockIdx.x * 128 + wave * 16;
    const float alpha = 1.0f + epsArr[layer];

    // ---- stage 0: A = alpha*h + agg from global ----
    for (int idx = t; idx < HD * HD; idx += 256) {
        int k = idx >> 7, n = idx & 127;
        wt[n * LDST + k] = Wl[idx];
    }
    __syncthreads();

    int arow = rowbase + m;
    if (arow >= nrows) arow = nrows - 1;
    const float* Ap  = h   + (size_t)arow * HD + 2 * khalf;
    const float* A2p = agg + (size_t)arow * HD + 2 * khalf;

    v8f acc[8];
    #pragma unroll
    for (int j = 0; j < 8; ++j) {
        float b = bl[j * 16 + m];
        #pragma unroll
        for (int r = 0; r < 8; ++r) acc[j][r] = b;
    }

    #pragma unroll 2
    for (int kk = 0; kk < 32; ++kk) {
        v2f a1 = *(const v2f*)(Ap  + kk * 4);
        v2f a2 = *(const v2f*)(A2p + kk * 4);
        v2f a;
        a.x = alpha * a1.x + a2.x;
        a.y = alpha * a1.y + a2.y;
        const int kb = kk * 4 + 2 * khalf;
        #pragma unroll
        for (int j = 0; j < 8; ++j) {
            v2f b = *(const v2f*)&wt[(j * 16 + m) * LDST + kb];
            acc[j] = __builtin_amdgcn_wmma_f32_16x16x4_f32(
                false, a, false, b, (short)0, acc[j], false, false);
        }
    }

    // relu + store tile to LDS (wave-private rows: no cross-wave hazard)
    float* atw = &at[(wave * 16 + khalf * 8) * LDST + m];
    #pragma unroll
    for (int j = 0; j < 8; ++j) {
        #pragma unroll
        for (int r = 0; r < 8; ++r)
            atw[r * LDST + j * 16] = fmaxf(acc[j][r], 0.0f);
    }

    // ---- stages 1..3: A from LDS tile ----
    const float* aLp = &at[(wave * 16 + m) * LDST + 2 * khalf];

    for (int s = 1; s < 4; ++s) {
        __syncthreads();                        // all wt reads of stage s-1 done
        const float* Ws = Wl + (size_t)s * HD * HD;
        for (int idx = t; idx < HD * HD; idx += 256) {
            int k = idx >> 7, n = idx & 127;
            wt[n * LDST + k] = Ws[idx];
        }
        __syncthreads();

        const float* bs = bl + s * HD;
        #pragma unroll
        for (int j = 0; j < 8; ++j) {
            float b = bs[j * 16 + m];
            #pragma unroll
            for (int r = 0; r < 8; ++r) acc[j][r] = b;
        }

        #pragma unroll 2
        for (int kk = 0; kk < 32; ++kk) {
            v2f a = *(const v2f*)(aLp + kk * 4);
            const int kb = kk * 4 + 2 * khalf;
            #pragma unroll
            for (int j = 0; j < 8; ++j) {
                v2f b = *(const v2f*)&wt[(j * 16 + m) * LDST + kb];
                acc[j] = __builtin_amdgcn_wmma_f32_16x16x4_f32(
                    false, a, false, b, (short)0, acc[j], false, false);
            }
        }

        if (s < 3) {
            // relu + write back to LDS tile (own rows only)
            #pragma unroll
            for (int j = 0; j < 8; ++j) {
                #pragma unroll
                for (int r = 0; r < 8; ++r)
                    atw[r * LDST + j * 16] = fmaxf(acc[j][r], 0.0f);
            }
        } else {
            // final stage: store Z to global, accumulate BN column stats
            float* Zp = Z + (size_t)(rowbase + khalf * 8) * HD + m;
            if (rowbase + 16 <= nrows) {        // uniform fast path
                #pragma unroll
                for (int j = 0; j < 8; ++j) {
                    float ssum = 0.0f, ssq = 0.0f;
                    #pragma unroll
                    for (int r = 0; r < 8; ++r) {
                        float v = acc[j][r];
                        Zp[(size_t)r * HD + j * 16] = v;
                        ssum += v;
                        ssq  += v * v;
                    }
                    unsafeAtomicAdd(&bnsum[j * 16 + m],   ssum);
                    unsafeAtomicAdd(&bnsumsq[j * 16 + m], ssq);
                }
            } else {
                #pragma unroll
                for (int j = 0; j < 8; ++j) {
                    float ssum = 0.0f, ssq = 0.0f;
                    #pragma unroll
                    for (int r = 0; r < 8; ++r) {
                        int row = rowbase + khalf * 8 + r;
                        float v = acc[j][r];
                        if (row < nrows) {
                            Zp[(size_t)r * HD + j * 16] = v;
                            ssum += v;
                            ssq  += v * v;
                        }
                    }
                    unsafeAtomicAdd(&bnsum[j * 16 + m],   ssum);
                    unsafeAtomicAdd(&bnsumsq[j * 16 + m], ssq);
                }
            }
        }
    }
}

// ---------------------------------------------------------------------------
// GINE message passing, edge projection recomputed from 16-d edge_attr:
//   msg = relu(h[src] + edge_attr @ edge_W + edge_b); agg[dst] += msg
// ---------------------------------------------------------------------------
__global__ __launch_bounds__(256) void message_kernel(
    const float* __restrict__ h, const float* __restrict__ edge_attr,
    const int* __restrict__ ei, const float* __restrict__ eW,
    const float* __restrict__ eb, float* __restrict__ agg, int nE)
{
    __shared__ float w[16 * HD];
    __shared__ float bsh[HD];
    const int t = threadIdx.x;
    for (int idx = t; idx < 16 * HD; idx += 256) w[idx] = eW[idx];
    if (t < 128) bsh[t] = eb[t];
    __syncthreads();

    const int wave = t >> 5;
    const int lane = t & 31;
    const int f    = lane * 4;
    const int base = blockIdx.x * 64 + wave * 8;

    for (int i = 0; i < 8; ++i) {
        const int e = base + i;
        if (e >= nE) break;
        const int src = ei[e];
        const int dst = ei[nE + e];

        const float4 av = *(const float4*)(edge_attr + (size_t)e * 16 + (lane & 3) * 4);
        float avf[4] = {av.x, av.y, av.z, av.w};

        float4 m4;
        m4.x = bsh[f]; m4.y = bsh[f + 1]; m4.z = bsh[f + 2]; m4.w = bsh[f + 3];
        #pragma unroll
        for (int d = 0; d < 16; ++d) {
            const float ad = __shfl(avf[d & 3], d >> 2, 32);
            const float4 wv = *(const float4*)&w[d * HD + f];
            m4.x = fmaf(ad, wv.x, m4.x);
            m4.y = fmaf(ad, wv.y, m4.y);
            m4.z = fmaf(ad, wv.z, m4.z);
            m4.w = fmaf(ad, wv.w, m4.w);
        }

        const float4 hs = *(const float4*)(h + (size_t)src * HD + f);
        float* ap = agg + (size_t)dst * HD + f;
        unsafeAtomicAdd(ap + 0, fmaxf(hs.x + m4.x, 0.0f));
        unsafeAtomicAdd(ap + 1, fmaxf(hs.y + m4.y, 0.0f));
        unsafeAtomicAdd(ap + 2, fmaxf(hs.z + m4.z, 0.0f));
        unsafeAtomicAdd(ap + 3, fmaxf(hs.w + m4.w, 0.0f));
    }
}

// ---------------------------------------------------------------------------
// BatchNorm finalize (+ReLU +residual): h = relu((z-mu)*rsqrt(var+eps)*g + b) + h
// ---------------------------------------------------------------------------
__global__ __launch_bounds__(256) void bn_res_kernel(
    const float* __restrict__ z, float* __restrict__ h,
    const float* __restrict__ bnsum, const float* __restrict__ bnsumsq,
    const float* __restrict__ gamma, const float* __restrict__ beta,
    int n, float invN)
{
    const int i = blockIdx.x * 256 + threadIdx.x;
    if (i >= n * HD) return;
    const int col = i & (HD - 1);
    const float mu  = bnsum[col] * invN;
    const float var = bnsumsq[col] * invN - mu * mu;
    const float inv = rsqrtf(var + BN_EPS);
    const float v = (z[i] - mu) * inv * gamma[col] + beta[col];
    h[i] = fmaxf(v, 0.0f) + h[i];
}

// ---------------------------------------------------------------------------
// Pooling: counts, atomic feature sums into d_out, then divide.
// ---------------------------------------------------------------------------
__global__ __launch_bounds__(256) void count_kernel(
    const int* __restrict__ batch, float* __restrict__ cnt, int n)
{
    const int i = blockIdx.x * 256 + threadIdx.x;
    if (i < n) unsafeAtomicAdd(&cnt[batch[i]], 1.0f);
}

__global__ __launch_bounds__(256) void pool_sum_kernel(
    const float* __restrict__ h, const int* __restrict__ batch,
    float* __restrict__ out, int n)
{
    const int i = blockIdx.x * 256 + threadIdx.x;   // n*32 threads
    const int node = i >> 5, lane = i & 31;
    if (node >= n) return;
    const int b = batch[node];
    const float4 v = *(const float4*)(h + (size_t)node * HD + lane * 4);
    float* op = out + (size_t)b * HD + lane * 4;
    unsafeAtomicAdd(op + 0, v.x);
    unsafeAtomicAdd(op + 1, v.y);
    unsafeAtomicAdd(op + 2, v.z);
    unsafeAtomicAdd(op + 3, v.w);
}

__global__ __launch_bounds__(256) void pool_div_kernel(
    float* __restrict__ out, const float* __restrict__ cnt, int s)
{
    const int i = blockIdx.x * 256 + threadIdx.x;
    if (i >= s * HD) return;
    out[i] /= fmaxf(cnt[i >> 7], 1.0f);
}

// ---------------------------------------------------------------------------
extern "C" void kernel_launch(void* const* d_in, const int* in_sizes, int n_in,
                              void* d_out, int out_size, void* d_ws, size_t ws_size,
                              hipStream_t stream)
{
    const float* x      = (const float*)d_in[0];
    const int*   ei     = (const int*)  d_in[1];
    const int*   batch  = (const int*)  d_in[2];
    const float* eattr  = (const float*)d_in[3];
    const float* nodeW  = (const float*)d_in[4];
    const float* nodeb  = (const float*)d_in[5];
    const float* edgeW  = (const float*)d_in[6];
    const float* edgeb  = (const float*)d_in[7];
    const float* eps    = (const float*)d_in[8];
    const float* mlpW   = (const float*)d_in[9];
    const float* mlpb   = (const float*)d_in[10];
    const float* gamma  = (const float*)d_in[11];
    const float* beta   = (const float*)d_in[12];
    float*       out    = (float*)d_out;

    const int N = in_sizes[0] / HD;      // 50000
    const int E = in_sizes[1] / 2;       // 600000
    const int S = out_size / HD;         // 1024
    const int L = 5, ML = 4;
    const size_t NH = (size_t)N * HD;

    float* ws      = (float*)d_ws;
    float* h       = ws;                 // [N,H]
    float* agg     = ws + NH;            // [N,H]
    float* zbuf    = ws + 2 * NH;        // [N,H]
    float* bnsum   = ws + 3 * NH;        // [128]
    float* bnsumsq = bnsum + HD;         // [128]
    float* cnt     = bnsumsq + HD;       // [S]

    const int gemmGrid = (N + 127) / 128;
    const int eltGrid  = (int)((NH + 255) / 256);

    // h = x @ node_W + node_b
    gemm_f32_wmma<<<gemmGrid, 256, 0, stream>>>(x, nodeW, nodeb, h, N);

    for (int l = 0; l < L; ++l) {
        hipMemsetAsync(agg, 0, NH * sizeof(float), stream);
        hipMemsetAsync(bnsum, 0, 2 * HD * sizeof(float), stream);

        message_kernel<<<(E + 63) / 64, 256, 0, stream>>>(
            h, eattr, ei, edgeW, edgeb, agg, E);

        // fused 4-layer MLP (+eps fusion, +BN stats)
        mlp4_f32_wmma<<<gemmGrid, 256, 0, stream>>>(
            h, agg, eps, l,
            mlpW + (size_t)(l * ML) * HD * HD, mlpb + (size_t)(l * ML) * HD,
            zbuf, N, bnsum, bnsumsq);

        bn_res_kernel<<<eltGrid, 256, 0, stream>>>(
            zbuf, h, bnsum, bnsumsq, gamma + l * HD, beta + l * HD, N,
            1.0f / (float)N);
    }

    // global mean pool over `batch`
    hipMemsetAsync(out, 0, (size_t)S * HD * sizeof(float), stream);
    hipMemsetAsync(cnt, 0, (size_t)S * sizeof(float), stream);
    count_kernel<<<(N + 255) / 256, 256, 0, stream>>>(batch, cnt, N);
    pool_sum_kernel<<<(N * 32 + 255) / 256, 256, 0, stream>>>(h, batch, out, N);
    pool_div_kernel<<<(S * HD + 255) / 256, 256, 0, stream>>>(out, cnt, S);
}